// LLamaSBmm_48361331752995
// MI455X (gfx1250) — compile-verified
//
#include <hip/hip_runtime.h>

// ---------------------------------------------------------------------------
// MoE 2:4-sparse int4 GEMM for MI455X (gfx1250, wave32).
// y[t,:] = x[t,:] @ W[e_t]^T,  W stored as packed int4 nonzeros + 2:4 meta.
// Sparse weights feed V_SWMMAC_F32_16X16X64_F16 directly (A = sparse W,
// B = dense x^T, SRC2 index = raw meta bits). Scales fold into the epilogue
// (groupsize == IN). Register-staged LDS ping-pong pipeline, one barrier/iter,
// b128 LDS traffic, b128 packed epilogue store.
// ---------------------------------------------------------------------------

typedef _Float16 half_t;
typedef _Float16 v2h  __attribute__((ext_vector_type(2)));
typedef _Float16 v8h  __attribute__((ext_vector_type(8)));
typedef _Float16 v16h __attribute__((ext_vector_type(16)));
typedef _Float16 v32h __attribute__((ext_vector_type(32)));
typedef float    v8f  __attribute__((ext_vector_type(8)));
typedef unsigned long long ull;

#define E_    8
#define IN_   4096
#define OUT_  4096
#define T_    8192
#define KCH_  (IN_ / 64)          // 64 K-chunks of 64 dense K each
#define MAXT_ (T_ / 16 + E_)      // 520 worst-case M tiles
#define XPITCH 72                 // x tile row pitch in halves (bank-conflict free)

// d_ws layout in int32 units (~40 KB total)
#define WS_CNT   0
#define WS_TILEC 8
#define WS_OFF   16
#define WS_FILL  24
#define WS_DESC  32
#define WS_TOK   2048

__global__ void k_zero(int* ws) {
  if (threadIdx.x < 32) ws[threadIdx.x] = 0;
}

__global__ void k_count(const int* __restrict__ idx, int* __restrict__ ws) {
  int t = blockIdx.x * 256 + threadIdx.x;
  atomicAdd(&ws[WS_CNT + idx[t]], 1);
}

__global__ void k_plan(int* ws) {
  if (threadIdx.x == 0 && blockIdx.x == 0) {
    int off = 0, tiles = 0;
    for (int e = 0; e < E_; ++e) {
      ws[WS_OFF + e] = off;
      ws[WS_FILL + e] = 0;
      int c = ws[WS_CNT + e];
      int nt = (c + 15) >> 4;
      for (int i = 0; i < nt; ++i) {
        ws[WS_DESC + 2 * tiles]     = e;
        ws[WS_DESC + 2 * tiles + 1] = i * 16;
        ++tiles;
      }
      off += c;
    }
    ws[WS_TILEC] = tiles;
  }
}

__global__ void k_scatter(const int* __restrict__ idx, int* __restrict__ ws) {
  int t = blockIdx.x * 256 + threadIdx.x;
  int e = idx[t];
  int p = atomicAdd(&ws[WS_FILL + e], 1);
  ws[WS_TOK + ws[WS_OFF + e] + p] = t;
}

// ---------------------------------------------------------------------------
// Main kernel: 256 threads = 8 waves. Block tile: 16 tokens x 128 outs;
// wave w owns the 16x16 tile D = W_sparse(16xK) * x^T(Kx16).
// ---------------------------------------------------------------------------
__global__ __launch_bounds__(256)
void k_moe_s4_gemm(const half_t* __restrict__ x,
                   const int*    __restrict__ qweight,
                   const short*  __restrict__ meta,
                   const half_t* __restrict__ scales,
                   const int*    __restrict__ ws,
                   half_t*       __restrict__ y)
{
  const int tileCount = ws[WS_TILEC];
  if ((int)blockIdx.x >= tileCount) return;

  const int e      = ws[WS_DESC + 2 * blockIdx.x];
  const int tb     = ws[WS_DESC + 2 * blockIdx.x + 1];
  const int cnt    = ws[WS_CNT + e];
  const int segoff = ws[WS_OFF + e];
  const int oBase  = blockIdx.y * 128;

  __shared__ __align__(16) half_t       lds_x[2][16 * XPITCH];
  __shared__ __align__(16) unsigned int lds_w[2][512];
  __shared__ __align__(16) ull          lds_m[2][128];
  __shared__ int lds_tok[16];

  const int tid  = threadIdx.x;
  const int wv   = tid >> 5;
  const int lane = tid & 31;
  const int M    = lane & 15;   // A row within wave tile; also B/D column
  const int hh   = lane >> 4;   // lane half
  const int ob   = oBase + wv * 16;

  if (tid < 16)
    lds_tok[tid] = (tb + tid < cnt) ? ws[WS_TOK + segoff + tb + tid] : -1;
  __syncthreads();

  const unsigned int* qw_e = (const unsigned int*)qweight + (size_t)e * (128 * 8192);
  const short*        mt_e = meta + (size_t)e * ((size_t)OUT_ * 256);

  // ---- chunk-invariant per-thread load setup -----------------------------
  // x gather: thread stages 8B of token row (tok = tid>>4, halves koff..koff+3)
  const int  tok    = tid >> 4;
  const int  koff   = (tid & 15) * 4;
  const int  tokid  = lds_tok[tok];
  const bool xvalid = tokid >= 0;
  const ull* xp = (const ull*)(x + (size_t)(xvalid ? tokid : 0) * IN_ + koff);
  // per chunk: +64 halves = +16 ull

  // qweight: thread stages words w = tid and tid+256 of the 512-word chunk.
  // word(c, o) = (c>>4)*8192 + (c&15)*512 + (o>>3);  c = kc*32 + cl
  int w0 = tid, w1 = tid + 256;
  int wv0 = w0 >> 6, r0 = w0 & 63, og0 = r0 & 1, cl0 = r0 >> 1;
  int wv1 = w1 >> 6, r1 = w1 & 63, og1 = r1 & 1, cl1 = r1 >> 1;
  const int l0 = wv0 * 64 + og0 * 32 + cl0;
  const int l1 = wv1 * 64 + og1 * 32 + cl1;
  const unsigned int* qp0 =
      qw_e + (cl0 >> 4) * 8192 + (cl0 & 15) * 512 + ((oBase + wv0 * 16) >> 3) + og0;
  const unsigned int* qp1 =
      qw_e + (cl1 >> 4) * 8192 + (cl1 & 15) * 512 + ((oBase + wv1 * 16) >> 3) + og1;
  // per chunk: +2*8192 words

  // meta: threads 0..127 stage row oBase+tid (4 int16 words per chunk)
  const ull* mp = (const ull*)(mt_e + (size_t)(oBase + (tid & 127)) * 256);
  const bool mvalid = tid < 128;   // wave-uniform

  // ---- loop-invariant epilogue data: 8 consecutive scales per lane -------
  union { uint4 q; v8h h; } S;
  S.q = *(const uint4*)(scales + (size_t)e * OUT_ + ob + 8 * hh);

  // ---- prime chunk 0 ------------------------------------------------------
  ull          xr = xp[0];
  unsigned int wr0 = qp0[0], wr1 = qp1[0];
  ull          mr = mvalid ? mp[0] : 0ull;
  if (!xvalid) xr = 0ull;

  v8f acc = {0.f, 0.f, 0.f, 0.f, 0.f, 0.f, 0.f, 0.f};
  const v2h bias = {(_Float16)1032.0f, (_Float16)1032.0f};  // 0x6400|nib = 1024+nib

  for (int kc = 0; kc < KCH_; ++kc) {
    const int b = kc & 1;
    // ---- commit staged registers to LDS buffer b ----
    *(ull*)(&lds_x[b][tok * XPITCH + koff]) = xr;
    lds_w[b][l0] = wr0;
    lds_w[b][l1] = wr1;
    if (mvalid) lds_m[b][tid] = mr;
    __syncthreads();

    // ---- issue chunk kc+1 global loads (fly during compute) ----
    if (kc + 1 < KCH_) {
      xr  = xp[(kc + 1) * 16];
      wr0 = qp0[(kc + 1) * 16384];
      wr1 = qp1[(kc + 1) * 16384];
      if (mvalid) mr = mp[kc + 1];
      if (!xvalid) xr = 0ull;
    }

    // ---- B fragment: dense x^T (ISA 7.12.4 layout), 4x ds_load_b128 ----
    union { v32h h; uint4 q[4]; } B;
    {
      const unsigned int* xw = (const unsigned int*)lds_x[b];
      const int b0 = M * (XPITCH / 2) + hh * 8;   // uint index of K=16*hh
      B.q[0] = *(const uint4*)(xw + b0);
      B.q[1] = *(const uint4*)(xw + b0 + 4);
      B.q[2] = *(const uint4*)(xw + b0 + 16);     // K = 32+16*hh
      B.q[3] = *(const uint4*)(xw + b0 + 20);
    }

    // ---- A fragment: sparse int4 -> f16 via mantissa-bias trick ----
    // lanes<16: stored c 0-7 (V0-3), 16-23 (V4-7); lanes>=16: 8-15, 24-31.
    union { v16h h; unsigned int u[8]; } A;
    {
      const unsigned int* wp = &lds_w[b][wv * 64 + (M >> 3) * 32 + hh * 8];
      unsigned int wa[16];
      *(uint4*)(wa + 0)  = *(const uint4*)(wp + 0);
      *(uint4*)(wa + 4)  = *(const uint4*)(wp + 4);
      *(uint4*)(wa + 8)  = *(const uint4*)(wp + 16);
      *(uint4*)(wa + 12) = *(const uint4*)(wp + 20);
      const int sh = 4 * (M & 7);
      #pragma unroll
      for (int j = 0; j < 8; ++j) {
        unsigned int n0 = (wa[2 * j]     >> sh) & 15u;
        unsigned int n1 = (wa[2 * j + 1] >> sh) & 15u;
        union { unsigned int u; v2h h; } p;
        p.u = 0x64006400u | n0 | (n1 << 16);      // {1024+n0, 1024+n1}
        p.h = p.h - bias;                          // v_pk_add 0xe408 -> {n0-8, n1-8}
        A.u[j] = p.u;
      }
    }

    // ---- SWMMAC index: raw meta bits (idx0<idx1 pairs per group-of-4) ----
    const int sidx = (int)(lds_m[b][wv * 16 + M] >> (hh * 32));

    acc = __builtin_amdgcn_swmmac_f32_16x16x64_f16(
        false, A.h, false, B.h, acc, sidx, false, false);
  }

  // ---- epilogue: lane owns 8 consecutive outs o = ob + 8*hh + (0..7) for
  // token lds_tok[M]; scale, pack to f16, single 16B store. ----
  {
    const int t = lds_tok[M];
    if (t >= 0) {
      union { v8h h; uint4 q; } O;
      #pragma unroll
      for (int r = 0; r < 8; ++r)
        O.h[r] = (half_t)(acc[r] * (float)S.h[r]);
      *(uint4*)(y + (size_t)t * OUT_ + ob + 8 * hh) = O.q;
    }
  }
}

// ---------------------------------------------------------------------------
extern "C" void kernel_launch(void* const* d_in, const int* in_sizes, int n_in,
                              void* d_out, int out_size, void* d_ws, size_t ws_size,
                              hipStream_t stream) {
  const half_t* x       = (const half_t*)d_in[0];
  const int*    indices = (const int*)   d_in[1];
  const int*    qweight = (const int*)   d_in[2];
  const short*  meta    = (const short*) d_in[3];
  const half_t* scales  = (const half_t*)d_in[4];
  half_t*       y       = (half_t*)d_out;
  int*          ws      = (int*)d_ws;

  k_zero   <<<1,        32,  0, stream>>>(ws);
  k_count  <<<T_ / 256, 256, 0, stream>>>(indices, ws);
  k_plan   <<<1,        1,   0, stream>>>(ws);
  k_scatter<<<T_ / 256, 256, 0, stream>>>(indices, ws);

  dim3 grid(MAXT_, OUT_ / 128, 1);
  k_moe_s4_gemm<<<grid, 256, 0, stream>>>(x, qweight, meta, scales, ws, y);
}